// Router_48653389529536
// MI455X (gfx1250) — compile-verified
//
#include <hip/hip_runtime.h>
#include <hip/hip_bf16.h>

typedef float v2f __attribute__((ext_vector_type(2)));
typedef float v4f __attribute__((ext_vector_type(4)));
typedef float v8f __attribute__((ext_vector_type(8)));

#define DIM   2048
#define SEQ   2048
#define BATCH 8
#define NEXP  8
#define WSEPS 1e-5f

// -------------------------------------------------------------------------
// Kernel A: xm[b,d] = mean over s of x[b,s,d].  Memory-bound: 134 MB read
// (~5.8 us at 23.3 TB/s HBM).  grid = (DIM/64, BATCH), 256 thr = 16 d-quads
// x 16 s-slices.  NT b128 streaming loads (read-once), prefetch ahead,
// deterministic LDS tree reduction (no float atomics).
// -------------------------------------------------------------------------
__global__ void __launch_bounds__(256)
seq_mean_kernel(const float* __restrict__ x, float* __restrict__ xm) {
    const int b     = blockIdx.y;
    const int dbase = blockIdx.x * 64;
    const int tid   = threadIdx.x;
    const int dq    = tid & 15;   // 16 quads of 4 floats = 64 columns
    const int ss    = tid >> 4;   // 16 s-slices

    const size_t stride = (size_t)16 * DIM;
    const float* p = x + (size_t)b * SEQ * DIM + (size_t)ss * DIM + dbase + dq * 4;

    v4f acc = {0.f, 0.f, 0.f, 0.f};
    #pragma unroll 4
    for (int it = 0; it < SEQ / 16; ++it) {
        __builtin_prefetch(p + 8 * stride, 0, 0);            // global_prefetch_b8
        v4f v = __builtin_nontemporal_load((const v4f*)p);   // NT: read-once stream
        acc += v;
        p += stride;
    }

    __shared__ v4f red[16][17];
    red[ss][dq] = acc;
    __syncthreads();
    #pragma unroll
    for (int w = 8; w > 0; w >>= 1) {
        if (ss < w) red[ss][dq] += red[ss + w][dq];
        __syncthreads();
    }
    if (ss == 0) {
        v4f r = red[0][dq] * (1.0f / SEQ);
        *(v4f*)(xm + (size_t)b * DIM + dbase + dq * 4) = r;
    }
}

// -------------------------------------------------------------------------
// Kernel B: weight-standardization stats, G = xm @ W^T via V_WMMA_F32_16X16X4,
// scores -> softmax -> top-k -> (combine, indices, topk_scores).
// Single block of 256 threads; wave 0 runs the branchless WMMA K-loop.
// -------------------------------------------------------------------------
__global__ void __launch_bounds__(256)
router_kernel(const float* __restrict__ W,
              const float* __restrict__ gain,
              const float* __restrict__ init_std,
              const float* __restrict__ noise,
              const int*   __restrict__ topk_p,
              const float* __restrict__ xm,
              float*       __restrict__ out) {
    const int tid = threadIdx.x;

    __shared__ float s_p1[NEXP][32];
    __shared__ float s_p2[NEXP][32];
    __shared__ float s_mean[NEXP];
    __shared__ float s_scale[NEXP];   // init_std * gain / (std + eps)
    __shared__ float s_sxm[BATCH];
    __shared__ float s_G[8][16];      // G[b][e] from WMMA accumulator

    // ---- Phase 1: per-expert mean / unbiased std of W rows -------------
    {
        const int e = tid >> 5, lane = tid & 31;
        const float* wp = W + e * DIM + lane * 64;
        float pw = 0.f, pw2 = 0.f;
        #pragma unroll 8
        for (int j = 0; j < 64; ++j) { float w = wp[j]; pw += w; pw2 += w * w; }
        s_p1[e][lane] = pw;
        s_p2[e][lane] = pw2;
    }
    __syncthreads();
    if (tid < NEXP) {
        float sw = 0.f, sw2 = 0.f;
        for (int l = 0; l < 32; ++l) { sw += s_p1[tid][l]; sw2 += s_p2[tid][l]; }
        float mean = sw / DIM;
        float var  = (sw2 - sw * sw / DIM) / (float)(DIM - 1);
        float sd   = sqrtf(fmaxf(var, 0.f)) + WSEPS;
        s_mean[tid]  = mean;
        s_scale[tid] = init_std[tid] * gain[tid] / sd;
    }
    __syncthreads();

    // ---- Phase 2: sxm[b] = sum_d xm[b,d] -------------------------------
    {
        const int b = tid >> 5, lane = tid & 31;
        const float* xp = xm + b * DIM + lane * 64;
        float px = 0.f;
        #pragma unroll 8
        for (int j = 0; j < 64; ++j) px += xp[j];
        s_p1[b][lane] = px;
    }
    __syncthreads();
    if (tid < BATCH) {
        float sx = 0.f;
        for (int l = 0; l < 32; ++l) sx += s_p1[tid][l];
        s_sxm[tid] = sx;
    }
    __syncthreads();

    // ---- Phase 3: G = xm(16x2048 pad) @ W^T(2048x16 pad), f32 WMMA -----
    // A 16x4: lane holds row M=lane%16, K-pair 2*(lane/16)+{0,1}.
    // B 4x16: lane holds col N=lane%16, same K-pair.
    // C 16x16: VGPR v, lanes 0-15 -> M=v, N=lane.
    // No masking needed: C[m][n] mixes only A-row m and B-col n, and we read
    // only m,n < 8 — so lanes with m>=8 alias row m&7 (finite data) harmlessly.
    // Keeps EXEC = all-ones through the whole loop (WMMA requirement) with a
    // branchless body: 2x global_load_b64 + 1x v_wmma per K-step.
    if (tid < 32) {
        const int lane = tid;
        const int m    = lane & 15;
        const int row  = m & 7;                 // alias for padding lanes
        const int kh   = (lane >> 4) << 1;      // K-pair base: 0 or 2
        const float* pa = xm + row * DIM + kh;
        const float* pb = W  + row * DIM + kh;
        v8f c = {};
        #pragma unroll 8
        for (int k0 = 0; k0 < DIM; k0 += 4) {
            v2f a  = *(const v2f*)(pa + k0);
            v2f b2 = *(const v2f*)(pb + k0);
            c = __builtin_amdgcn_wmma_f32_16x16x4_f32(
                    /*neg_a=*/false, a, /*neg_b=*/false, b2,
                    /*c_mod=*/(short)0, c, /*reuse_a=*/false, /*reuse_b=*/false);
        }
        if (lane < 16) {
            #pragma unroll
            for (int v = 0; v < 8; ++v) s_G[v][lane] = c[v];  // rows M=0..7
        }
    }
    __syncthreads();

    // ---- Phase 4: scores -> softmax -> top-k ---------------------------
    if (tid < BATCH) {
        const int b = tid;
        int k = topk_p[0];
        if (k < 1) k = 1;
        if (k > NEXP) k = NEXP;

        float sc[NEXP];
        float mx = -3.4e38f;
        #pragma unroll
        for (int e = 0; e < NEXP; ++e) {
            float s = s_scale[e] * (s_G[b][e] - s_mean[e] * s_sxm[b]) + noise[b * NEXP + e];
            sc[e] = s;
            mx = fmaxf(mx, s);
        }
        float den = 0.f;
        #pragma unroll
        for (int e = 0; e < NEXP; ++e) { sc[e] = __expf(sc[e] - mx); den += sc[e]; }
        float inv = 1.0f / den;
        #pragma unroll
        for (int e = 0; e < NEXP; ++e) sc[e] *= inv;   // gate probs

        float comb[NEXP];
        bool  used[NEXP];
        #pragma unroll
        for (int e = 0; e < NEXP; ++e) { comb[e] = 0.f; used[e] = false; }

        // out layout: combine[8][8] @0, indices[8][k] @64, topk @64+8k
        for (int j = 0; j < k; ++j) {
            int best = 0; float bv = -1.f;
            for (int e = 0; e < NEXP; ++e)
                if (!used[e] && sc[e] > bv) { bv = sc[e]; best = e; }
            used[best] = true;
            comb[best] = bv;
            out[64 + b * k + j]             = (float)best;  // indices (as f32)
            out[64 + BATCH * k + b * k + j] = bv;           // topk gate scores
        }
        #pragma unroll
        for (int e = 0; e < NEXP; ++e) out[b * NEXP + e] = comb[e];
    }
}

// -------------------------------------------------------------------------
extern "C" void kernel_launch(void* const* d_in, const int* in_sizes, int n_in,
                              void* d_out, int out_size, void* d_ws, size_t ws_size,
                              hipStream_t stream) {
    const float* x      = (const float*)d_in[0];   // [8,2048,2048]
    const float* W      = (const float*)d_in[1];   // [8,2048]
    const float* gain   = (const float*)d_in[2];   // [8]
    const float* istd   = (const float*)d_in[3];   // [8]
    const float* noise  = (const float*)d_in[4];   // [8,8]
    const int*   topk   = (const int*)d_in[5];     // scalar

    float* out = (float*)d_out;
    float* xm  = (float*)d_ws;                     // [8,2048] = 64 KB scratch

    dim3 gridA(DIM / 64, BATCH);                   // 32 x 8 = 256 blocks
    seq_mean_kernel<<<gridA, 256, 0, stream>>>(x, xm);
    router_kernel<<<1, 256, 0, stream>>>(W, gain, istd, noise, topk, xm, out);
}